// FNO1d_58755152609654
// MI455X (gfx1250) — compile-verified
//
#include <hip/hip_runtime.h>
#include <hip/hip_bf16.h>
#include <math.h>

typedef __attribute__((ext_vector_type(2))) float v2f;
typedef __attribute__((ext_vector_type(8))) float v8f;

#define S   64
#define SS  4096
#define C   16
#define NB  4
#define M1  4
#define M2  5
#define PI_F 3.14159265358979323846f

#if defined(__gfx1250__) && __has_builtin(__builtin_amdgcn_wmma_f32_16x16x4_f32)
#define HAS_WMMA_F32 1
#endif

__device__ inline float2 cmul(float2 a, float2 b) {
  return make_float2(a.x * b.x - a.y * b.y, a.x * b.y + a.y * b.x);
}
__device__ inline float2 cadd(float2 a, float2 b) {
  return make_float2(a.x + b.x, a.y + b.y);
}

// ---------------------------------------------------------------- fc0 + NHWC->NCHW
__global__ void k_fc0(const float* __restrict__ x, const float* __restrict__ w,
                      const float* __restrict__ b, float* __restrict__ h) {
  int t = blockIdx.x * blockDim.x + threadIdx.x;  // 0..16383
  int bidx = t >> 12;
  int pix = t & 4095;
  int i = pix >> 6, j = pix & 63;
  float xv = x[t];
  float gx = (float)i * (1.0f / 63.0f);
  float gy = (float)j * (1.0f / 63.0f);
  for (int c = 0; c < C; ++c) {
    h[((bidx * C + c) << 12) + pix] =
        w[c * 3 + 0] * xv + w[c * 3 + 1] * gx + w[c * 3 + 2] * gy + b[c];
  }
}

// ---------------------------------------------------------------- instance norm (per b,c)
__global__ void k_inorm(const float* __restrict__ src, float* __restrict__ dst) {
  int bc = blockIdx.x;
  const float* p = src + ((size_t)bc << 12);
  float* q = dst + ((size_t)bc << 12);
  int t = threadIdx.x;
  float s = 0.f, s2 = 0.f;
  for (int e = t; e < SS; e += 256) { float v = p[e]; s += v; s2 += v * v; }
  __shared__ float rs[256], rq[256];
  rs[t] = s; rq[t] = s2;
  __syncthreads();
  for (int off = 128; off > 0; off >>= 1) {
    if (t < off) { rs[t] += rs[t + off]; rq[t] += rq[t + off]; }
    __syncthreads();
  }
  float mean = rs[0] * (1.0f / SS);
  float var = rq[0] * (1.0f / SS) - mean * mean;
  float inv = rsqrtf(var + 1e-5f);
  for (int e = t; e < SS; e += 256) q[e] = (p[e] - mean) * inv;
}

// ---------------------------------------------------------------- forward 2D DFT (real -> complex)
__global__ void k_fft2(const float* __restrict__ hn, float2* __restrict__ alpha) {
  int bc = blockIdx.x;
  __shared__ float xr[SS];
  __shared__ float2 t1[SS];
  __shared__ float2 tw[S];
  int t = threadIdx.x;
  const float* p = hn + ((size_t)bc << 12);
  for (int e = t; e < SS; e += 256) xr[e] = p[e];
  if (t < S) {
    float ang = -2.0f * PI_F * (float)t / (float)S;
    tw[t] = make_float2(cosf(ang), sinf(ang));
  }
  __syncthreads();
  // rows (axis -1): t1[i][k2] = sum_j xr[i][j] * tw[(j*k2)&63]
  for (int e = t; e < SS; e += 256) {
    int i = e >> 6, k2 = e & 63;
    float2 acc = make_float2(0.f, 0.f);
    const float* row = xr + (i << 6);
    for (int j = 0; j < S; ++j) {
      float2 wv = tw[(j * k2) & 63];
      acc.x += row[j] * wv.x;
      acc.y += row[j] * wv.y;
    }
    t1[e] = acc;
  }
  __syncthreads();
  // columns (axis -2): alpha[k1][k2] = sum_i t1[i][k2] * tw[(i*k1)&63]
  float2* out = alpha + ((size_t)bc << 12);
  for (int e = t; e < SS; e += 256) {
    int k1 = e >> 6, k2 = e & 63;
    float2 acc = make_float2(0.f, 0.f);
    for (int i = 0; i < S; ++i)
      acc = cadd(acc, cmul(t1[(i << 6) + k2], tw[(i * k1) & 63]));
    out[e] = acc;
  }
}

// ---------------------------------------------------------------- pole reciprocals + A2 = sum_q res/d2
__global__ void k_poles(const float* __restrict__ p1r, const float* __restrict__ p1i,
                        const float* __restrict__ p2r, const float* __restrict__ p2i,
                        const float* __restrict__ resr, const float* __restrict__ resi,
                        const float* __restrict__ tg, const float* __restrict__ xg,
                        float2* __restrict__ invd1, float2* __restrict__ invd2,
                        float2* __restrict__ A2c) {
  int ik = blockIdx.x;      // i*16+k
  int t = threadIdx.x;      // 64: frequency index o / x
  int kk = (t < 32) ? t : t - 64;
  float dty = xg[1] - xg[0];
  float dtx = tg[1] - tg[0];
  float om1 = 2.0f * PI_F * (float)kk / ((float)S * dty);
  float om2 = 2.0f * PI_F * (float)kk / ((float)S * dtx);
  for (int p = 0; p < M1; ++p) {
    float ar = -p1r[ik * M1 + p];
    float ai = om1 - p1i[ik * M1 + p];
    float inv = 1.0f / (ar * ar + ai * ai);
    invd1[(ik * S + t) * M1 + p] = make_float2(ar * inv, -ai * inv);
  }
  float2 d2[M2];
  for (int q = 0; q < M2; ++q) {
    float ar = -p2r[ik * M2 + q];
    float ai = om2 - p2i[ik * M2 + q];
    float inv = 1.0f / (ar * ar + ai * ai);
    d2[q] = make_float2(ar * inv, -ai * inv);
    invd2[(ik * S + t) * M2 + q] = d2[q];
  }
  for (int p = 0; p < M1; ++p) {
    float2 acc = make_float2(0.f, 0.f);
    for (int q = 0; q < M2; ++q) {
      float2 rc = make_float2(resr[ik * 20 + p * M2 + q], resi[ik * 20 + p * M2 + q]);
      acc = cadd(acc, cmul(rc, d2[q]));
    }
    A2c[(ik * S + t) * M1 + p] = acc;
  }
}

// ---------------------------------------------------------------- r1[b,k,o,x] = sum_i alpha * Hsum
__global__ void k_r1(const float2* __restrict__ alpha, const float2* __restrict__ invd1,
                     const float2* __restrict__ A2c, float2* __restrict__ r1) {
  int bk = blockIdx.x;
  int b = bk >> 4, k = bk & 15;
  int t = threadIdx.x;  // 256
  __shared__ float2 l1[S * M1];  // invd1[o][p]
  __shared__ float2 l2[S * M1];  // A2[x][p]
  float2 acc[16];
  for (int n = 0; n < 16; ++n) acc[n] = make_float2(0.f, 0.f);
  for (int i = 0; i < C; ++i) {
    int ik = i * 16 + k;
    l1[t] = invd1[(size_t)ik * S * M1 + t];
    l2[t] = A2c[(size_t)ik * S * M1 + t];
    __syncthreads();
    const float2* al = alpha + ((size_t)(b * C + i) << 12);
    for (int n = 0; n < 16; ++n) {
      int e = t + (n << 8);
      int o = e >> 6, x = e & 63;
      float2 hs = make_float2(0.f, 0.f);
      for (int p = 0; p < M1; ++p)
        hs = cadd(hs, cmul(l1[o * M1 + p], l2[x * M1 + p]));
      acc[n] = cadd(acc[n], cmul(al[e], hs));
    }
    __syncthreads();
  }
  float2* out = r1 + ((size_t)bk << 12);
  for (int n = 0; n < 16; ++n) out[t + (n << 8)] = acc[n];
}

// ---------------------------------------------------------------- r2 partials per (b,i,k)
__global__ void k_r2p(const float2* __restrict__ alpha, const float2* __restrict__ invd1,
                      const float2* __restrict__ invd2, const float* __restrict__ resr,
                      const float* __restrict__ resi, float2* __restrict__ Vres) {
  int blk = blockIdx.x;                 // b*256 + i*16 + k
  int b = blk >> 8, ik = blk & 255, i = ik >> 4;
  int t = threadIdx.x;                  // 64 : o index
  __shared__ float2 ld2[S * M2];
  __shared__ float2 ld1[S * M1];
  __shared__ float2 Tl[S * M2];
  for (int e = t; e < S * M2; e += 64) ld2[e] = invd2[(size_t)ik * S * M2 + e];
  for (int e = t; e < S * M1; e += 64) ld1[e] = invd1[(size_t)ik * S * M1 + e];
  __syncthreads();
  const float2* al = alpha + ((size_t)(b * C + i) << 12) + (t << 6);  // row o=t
  float2 T[M2];
  for (int q = 0; q < M2; ++q) T[q] = make_float2(0.f, 0.f);
  for (int x = 0; x < S; ++x) {
    float2 a = al[x];
    for (int q = 0; q < M2; ++q) T[q] = cadd(T[q], cmul(a, ld2[x * M2 + q]));
  }
  for (int q = 0; q < M2; ++q) Tl[t * M2 + q] = T[q];
  __syncthreads();
  if (t < 20) {
    int p = t / M2, q = t % M2;
    float2 V = make_float2(0.f, 0.f);
    for (int o = 0; o < S; ++o) V = cadd(V, cmul(ld1[o * M1 + p], Tl[o * M2 + q]));
    float2 rc = make_float2(resr[ik * 20 + t], resi[ik * 20 + t]);
    Vres[(size_t)blk * 20 + t] = cmul(rc, V);
  }
}

// ---------------------------------------------------------------- deterministic r2 reduce over i
__global__ void k_r2red(const float2* __restrict__ Vres, float2* __restrict__ r2) {
  int e = blockIdx.x * 256 + threadIdx.x;
  if (e >= NB * C * 20) return;
  int pq = e % 20;
  int bk = e / 20;
  int b = bk >> 4, k = bk & 15;
  float2 acc = make_float2(0.f, 0.f);
  for (int i = 0; i < C; ++i)
    acc = cadd(acc, Vres[((size_t)(b * C + i) * C + k) * 20 + pq]);
  r2[e] = acc;  // [(b*16+k)*20 + pq]
}

// ---------------------------------------------------------------- inverse DFT, column pass (axis -2)
__global__ void k_ifft_col(const float2* __restrict__ r1, float2* __restrict__ scr) {
  int bc = blockIdx.x;
  int t = threadIdx.x;
  __shared__ float2 ld[SS];
  __shared__ float2 tw[S];
  const float2* p = r1 + ((size_t)bc << 12);
  for (int e = t; e < SS; e += 256) ld[e] = p[e];
  if (t < S) {
    float ang = 2.0f * PI_F * (float)t / (float)S;
    tw[t] = make_float2(cosf(ang), sinf(ang));
  }
  __syncthreads();
  float2* out = scr + ((size_t)bc << 12);
  for (int e = t; e < SS; e += 256) {
    int z = e >> 6, x = e & 63;
    float2 acc = make_float2(0.f, 0.f);
    for (int o = 0; o < S; ++o)
      acc = cadd(acc, cmul(ld[(o << 6) + x], tw[(o * z) & 63]));
    out[e] = acc;
  }
}

// ---------------------------------------------------------------- inverse DFT, row pass -> real, *1/4096
__global__ void k_ifft_row(const float2* __restrict__ scr, float* __restrict__ sout) {
  int bc = blockIdx.x;
  int t = threadIdx.x;
  __shared__ float2 ld[SS];
  __shared__ float2 tw[S];
  const float2* p = scr + ((size_t)bc << 12);
  for (int e = t; e < SS; e += 256) ld[e] = p[e];
  if (t < S) {
    float ang = 2.0f * PI_F * (float)t / (float)S;
    tw[t] = make_float2(cosf(ang), sinf(ang));
  }
  __syncthreads();
  float* out = sout + ((size_t)bc << 12);
  for (int e = t; e < SS; e += 256) {
    int z = e >> 6, y = e & 63;
    float acc = 0.f;
    const float2* row = ld + (z << 6);
    for (int x = 0; x < S; ++x) {
      float2 wv = tw[(x * y) & 63];
      acc += row[x].x * wv.x - row[x].y * wv.y;
    }
    out[e] = acc * (1.0f / SS);
  }
}

// ---------------------------------------------------------------- term3: sout += Re(sum r2*e1*e2)/4096
__global__ void k_x2(const float2* __restrict__ r2, const float* __restrict__ p1r,
                     const float* __restrict__ p1i, const float* __restrict__ p2r,
                     const float* __restrict__ p2i, const float* __restrict__ tg,
                     const float* __restrict__ xg, float* __restrict__ sout) {
  int blk = blockIdx.x;  // b*16 + out-channel io
  int b = blk >> 4, io = blk & 15;
  int t = threadIdx.x;  // 256
  __shared__ float2 e1[M1 * S];
  __shared__ float2 e2[M2 * S];
  __shared__ float2 r2c[20];
  __shared__ float2 G[M2 * S];
  float acc[16];
  for (int n = 0; n < 16; ++n) acc[n] = 0.f;
  for (int c = 0; c < C; ++c) {
    int ci = c * 16 + io;
    {  // e1[p][z] = exp(p1[c,io,p] * xg[z])
      int p = t >> 6, z = t & 63;
      float ty = xg[z];
      float ar = p1r[ci * M1 + p] * ty, ai = p1i[ci * M1 + p] * ty;
      float ex = expf(ar);
      e1[t] = make_float2(ex * cosf(ai), ex * sinf(ai));
    }
    for (int idx = t; idx < M2 * S; idx += 256) {  // e2[q][y] = exp(p2 * tg[y])
      int q = idx >> 6, y = idx & 63;
      float tx = tg[y];
      float ar = p2r[ci * M2 + q] * tx, ai = p2i[ci * M2 + q] * tx;
      float ex = expf(ar);
      e2[idx] = make_float2(ex * cosf(ai), ex * sinf(ai));
    }
    if (t < 20) r2c[t] = r2[(b * C + c) * 20 + t];
    __syncthreads();
    for (int idx = t; idx < M2 * S; idx += 256) {  // G[q][z] = sum_p r2c[p][q]*e1[p][z]
      int q = idx >> 6, z = idx & 63;
      float2 g = make_float2(0.f, 0.f);
      for (int p = 0; p < M1; ++p)
        g = cadd(g, cmul(r2c[p * M2 + q], e1[(p << 6) + z]));
      G[idx] = g;
    }
    __syncthreads();
    for (int n = 0; n < 16; ++n) {
      int e = t + (n << 8);
      int z = e >> 6, y = e & 63;
      float s = 0.f;
      for (int q = 0; q < M2; ++q) {
        float2 g = G[(q << 6) + z];
        float2 w = e2[(q << 6) + y];
        s += g.x * w.x - g.y * w.y;
      }
      acc[n] += s;
    }
    __syncthreads();
  }
  float* out = sout + ((size_t)blk << 12);
  for (int n = 0; n < 16; ++n) {
    int e = t + (n << 8);
    out[e] += acc[n] * (1.0f / SS);
  }
}

// ---------------------------------------------------------------- WMMA 1x1 conv (w0) + bias + residual
// D(16x16) = w0(16x16 [o][i]) x h_tile(16x16 [i][pix]); out pixel-major for fc1.
__global__ void k_w0(const float* __restrict__ h, const float* __restrict__ w0w,
                     const float* __restrict__ w0b, const float* __restrict__ soutn,
                     float* __restrict__ hfin) {
  int blk = blockIdx.x;            // b*256 + tile
  int b = blk >> 8, tile = blk & 255;
  int pixbase = tile << 4;
  int lane = threadIdx.x, lo = lane & 15, hi = lane >> 4;
  __shared__ float At[16 * 17];
  __shared__ float Bt[16 * 17];
  for (int idx = lane; idx < 256; idx += 32) {
    At[(idx >> 4) * 17 + (idx & 15)] = w0w[idx];  // A[o][i]
  }
  for (int idx = lane; idx < 256; idx += 32) {
    int i = idx >> 4, n = idx & 15;
    Bt[i * 17 + n] = h[((size_t)(b * C + i) << 12) + pixbase + n];  // B[i][pix]
  }
  __syncthreads();
  v8f cacc = {0.f, 0.f, 0.f, 0.f, 0.f, 0.f, 0.f, 0.f};
#ifdef HAS_WMMA_F32
  for (int k0 = 0; k0 < 16; k0 += 4) {
    v2f a, bv;
    a[0] = At[lo * 17 + k0 + hi * 2];
    a[1] = At[lo * 17 + k0 + hi * 2 + 1];
    bv[0] = Bt[(k0 + hi * 2) * 17 + lo];
    bv[1] = Bt[(k0 + hi * 2 + 1) * 17 + lo];
    cacc = __builtin_amdgcn_wmma_f32_16x16x4_f32(false, a, false, bv, (short)0,
                                                 cacc, false, false);
  }
#else
  for (int r = 0; r < 8; ++r) {
    int m = r + 8 * hi;
    float s = 0.f;
    for (int kx = 0; kx < 16; ++kx) s += At[m * 17 + kx] * Bt[kx * 17 + lo];
    cacc[r] = s;
  }
#endif
  for (int r = 0; r < 8; ++r) {
    int o = r + 8 * hi;
    int pix = pixbase + lo;
    float val = cacc[r] + w0b[o] + soutn[((size_t)(b * C + o) << 12) + pix];
    hfin[(size_t)(b * SS + pix) * C + o] = val;
  }
}

// ---------------------------------------------------------------- WMMA fc1 + sin + fc2 dot -> out
__global__ void k_fc12(const float* __restrict__ hfin, const float* __restrict__ fc1w,
                       const float* __restrict__ fc1b, const float* __restrict__ fc2w,
                       const float* __restrict__ fc2b, float* __restrict__ out) {
  int tile = blockIdx.x;           // 0..1023, 16 pixels each
  int pixbase = tile << 4;
  int lane = threadIdx.x, lo = lane & 15, hi = lane >> 4;
  __shared__ float At[16 * 17];
  __shared__ float Bt[16 * 17];
  __shared__ float red[32 * 8];
  for (int idx = lane; idx < 256; idx += 32) {
    int m = idx >> 4, kx = idx & 15;
    At[m * 17 + kx] = hfin[(size_t)(pixbase + m) * C + kx];  // A[pix][k]
  }
  float partial[8];
  for (int r = 0; r < 8; ++r) partial[r] = 0.f;
  for (int nt = 0; nt < 8; ++nt) {
    __syncthreads();
    for (int idx = lane; idx < 256; idx += 32) {
      int n = idx >> 4, kx = idx & 15;
      Bt[kx * 17 + n] = fc1w[(size_t)(nt * 16 + n) * C + kx];  // B[k][n]
    }
    __syncthreads();
    v8f cacc = {0.f, 0.f, 0.f, 0.f, 0.f, 0.f, 0.f, 0.f};
#ifdef HAS_WMMA_F32
    for (int k0 = 0; k0 < 16; k0 += 4) {
      v2f a, bv;
      a[0] = At[lo * 17 + k0 + hi * 2];
      a[1] = At[lo * 17 + k0 + hi * 2 + 1];
      bv[0] = Bt[(k0 + hi * 2) * 17 + lo];
      bv[1] = Bt[(k0 + hi * 2 + 1) * 17 + lo];
      cacc = __builtin_amdgcn_wmma_f32_16x16x4_f32(false, a, false, bv, (short)0,
                                                   cacc, false, false);
    }
#else
    for (int r = 0; r < 8; ++r) {
      int m = r + 8 * hi;
      float s = 0.f;
      for (int kx = 0; kx < 16; ++kx) s += At[m * 17 + kx] * Bt[kx * 17 + lo];
      cacc[r] = s;
    }
#endif
    int n = nt * 16 + lo;
    float fw = fc2w[n];
    float fb = fc1b[n];
    for (int r = 0; r < 8; ++r) {
      partial[r] += sinf(cacc[r] + fb) * fw;  // D element (M=r+8*hi, N=lo)
    }
  }
  for (int r = 0; r < 8; ++r) red[lane * 8 + r] = partial[r];
  __syncthreads();
  if (lane < 16) {
    int m = lane;
    int r = m & 7, hi2 = m >> 3;
    float s = 0.f;
    for (int src = hi2 * 16; src < hi2 * 16 + 16; ++src) s += red[src * 8 + r];
    out[pixbase + m] = s + fc2b[0];
  }
}

// ---------------------------------------------------------------- host launcher
extern "C" void kernel_launch(void* const* d_in, const int* in_sizes, int n_in,
                              void* d_out, int out_size, void* d_ws, size_t ws_size,
                              hipStream_t stream) {
  (void)in_sizes; (void)n_in; (void)out_size; (void)ws_size;
  const float* x    = (const float*)d_in[0];
  const float* fc0w = (const float*)d_in[1];
  const float* fc0b = (const float*)d_in[2];
  const float* p1r  = (const float*)d_in[3];
  const float* p1i  = (const float*)d_in[4];
  const float* p2r  = (const float*)d_in[5];
  const float* p2i  = (const float*)d_in[6];
  const float* resr = (const float*)d_in[7];
  const float* resi = (const float*)d_in[8];
  const float* w0w  = (const float*)d_in[9];
  const float* w0b  = (const float*)d_in[10];
  const float* fc1w = (const float*)d_in[11];
  const float* fc1b = (const float*)d_in[12];
  const float* fc2w = (const float*)d_in[13];
  const float* fc2b = (const float*)d_in[14];
  const float* tg   = (const float*)d_in[15];
  const float* xg   = (const float*)d_in[16];
  float* out = (float*)d_out;

  char* wp = (char*)d_ws;
  auto alloc = [&](size_t bytes) -> char* {
    char* p = wp;
    wp += (bytes + 255) & ~((size_t)255);
    return p;
  };
  float*  h     = (float*) alloc((size_t)262144 * 4);
  float*  hn    = (float*) alloc((size_t)262144 * 4);
  float2* alpha = (float2*)alloc((size_t)262144 * 8);
  float2* invd1 = (float2*)alloc((size_t)65536 * 8);
  float2* invd2 = (float2*)alloc((size_t)81920 * 8);
  float2* A2c   = (float2*)alloc((size_t)65536 * 8);
  float2* r1    = (float2*)alloc((size_t)262144 * 8);
  float2* scr   = (float2*)alloc((size_t)262144 * 8);
  float2* Vres  = (float2*)alloc((size_t)20480 * 8);
  float2* r2    = (float2*)alloc((size_t)1280 * 8);
  float*  sout  = (float*) alloc((size_t)262144 * 4);
  float*  soutn = (float*) alloc((size_t)262144 * 4);
  float*  hfin  = (float*) alloc((size_t)262144 * 4);

  k_fc0<<<64, 256, 0, stream>>>(x, fc0w, fc0b, h);
  k_inorm<<<64, 256, 0, stream>>>(h, hn);
  k_fft2<<<64, 256, 0, stream>>>(hn, alpha);
  k_poles<<<256, 64, 0, stream>>>(p1r, p1i, p2r, p2i, resr, resi, tg, xg,
                                  invd1, invd2, A2c);
  k_r1<<<64, 256, 0, stream>>>(alpha, invd1, A2c, r1);
  k_r2p<<<1024, 64, 0, stream>>>(alpha, invd1, invd2, resr, resi, Vres);
  k_r2red<<<5, 256, 0, stream>>>(Vres, r2);
  k_ifft_col<<<64, 256, 0, stream>>>(r1, scr);
  k_ifft_row<<<64, 256, 0, stream>>>(scr, sout);
  k_x2<<<64, 256, 0, stream>>>(r2, p1r, p1i, p2r, p2i, tg, xg, sout);
  k_inorm<<<64, 256, 0, stream>>>(sout, soutn);
  k_w0<<<1024, 32, 0, stream>>>(h, w0w, w0b, soutn, hfin);
  k_fc12<<<1024, 32, 0, stream>>>(hfin, fc1w, fc1b, fc2w, fc2b, out);
}